// JacobiGNN_27908697489845
// MI455X (gfx1250) — compile-verified
//
#include <hip/hip_runtime.h>
#include <cstdint>

typedef __attribute__((ext_vector_type(2))) float v2f;
typedef __attribute__((ext_vector_type(8))) float v8f;

#define N_NODES 8192
#define F_INF   512
#define HIDF    64
#define CCH     16
#define KORD    10
#define JCHUNK  512               // K-split chunk for both big GEMMs
#define NCHUNKS (N_NODES / JCHUNK)  // 16

__device__ __forceinline__ v8f wmma4(v2f a, v2f b, v8f c) {
  // V_WMMA_F32_16X16X4_F32: D(16x16 f32) = A(16x4 f32) * B(4x16 f32) + C
  return __builtin_amdgcn_wmma_f32_16x16x4_f32(
      /*neg_a=*/false, a, /*neg_b=*/false, b,
      /*c_mod=*/(short)0, c, /*reuse_a=*/false, /*reuse_b=*/false);
}

// ---------------------------------------------------------------------------
// Kernel 0: zero the two fp32 accumulator buffers (tbuf + obuf, contiguous).
// ---------------------------------------------------------------------------
__global__ void zero_kernel(float4* __restrict__ p) {
  p[blockIdx.x * blockDim.x + threadIdx.x] = make_float4(0.f, 0.f, 0.f, 0.f);
}

// ---------------------------------------------------------------------------
// Kernel 1: Jacobi polynomial filter.  Hf[n*16 + c]  (== reference Hfilt.T)
// ---------------------------------------------------------------------------
__global__ void jacobi_kernel(const float* __restrict__ origin_e,
                              const float* __restrict__ temp,
                              float* __restrict__ Hf) {
  __shared__ float al[CCH * (KORD + 1)];
  int tid = threadIdx.x;
  if (tid < CCH * (KORD + 1)) al[tid] = 0.5f * tanhf(temp[tid]);
  __syncthreads();

  int n = blockIdx.x * blockDim.x + tid;
  if (n >= N_NODES) return;
  float e = origin_e[n];

  const float a = 1.0f, b = 1.0f, lo = -1.0f, hi = 1.0f;
  const float coef1 = (a - b) * 0.5f - (a + b + 2.0f) * 0.5f * (lo + hi) / (hi - lo);
  const float coef2 = (a + b + 2.0f) / (hi - lo);

  for (int c = 0; c < CCH; ++c) {
    const float* ac = &al[c * (KORD + 1)];
    float xs_m2 = ac[0];
    float acc = xs_m2;
    float xs_m1 = coef1 * ac[1] + coef2 * ac[1] * e * xs_m2;
    acc += xs_m1;
#pragma unroll
    for (int L = 2; L <= KORD; ++L) {
      float Lf = (float)L;
      float coef_l  = 2.0f * Lf * (Lf + a + b) * (2.0f * Lf - 2.0f + a + b);
      float c_lm1_1 = (2.0f * Lf + a + b - 1.0f) * (2.0f * Lf + a + b) * (2.0f * Lf + a + b - 2.0f);
      float c_lm1_2 = (2.0f * Lf + a + b - 1.0f) * (a * a - b * b);
      float c_lm2   = 2.0f * (Lf - 1.0f + a) * (Lf - 1.0f + b) * (2.0f * Lf + a + b);
      float tmp1   = ac[L - 1] * (c_lm1_1 / coef_l);
      float tmp2   = ac[L - 1] * (c_lm1_2 / coef_l);
      float tmp3   = ac[L - 1] * ac[L - 2] * (c_lm2 / coef_l);
      float tmp1_2 = tmp1 * (2.0f / (hi - lo));
      float tmp2_2 = tmp1 * ((hi + lo) / (hi - lo)) + tmp2;
      float nx = tmp1_2 * (e * xs_m1) - tmp2_2 * xs_m1 - tmp3 * xs_m2;
      acc += nx;
      xs_m2 = xs_m1;
      xs_m1 = nx;
    }
    Hf[n * CCH + c] = acc;
  }
}

// ---------------------------------------------------------------------------
// Kernel 2: fused MLP  z = relu(x@W1 + b1)@W2 + b2   -> zb [8192,16]
// block = 128 threads (4 waves), one 16-row tile of x per block.
// ---------------------------------------------------------------------------
#define XS_STRIDE 516   // 512 + 4 pad -> conflict-free A reads
#define HS_STRIDE 68    // 64 + 4 pad

__global__ void mlp_kernel(const float* __restrict__ x,
                           const float* __restrict__ W1,
                           const float* __restrict__ b1,
                           const float* __restrict__ W2,
                           const float* __restrict__ b2,
                           float* __restrict__ zb) {
  __shared__ float xs[16 * XS_STRIDE];
  __shared__ float hs[16 * HS_STRIDE];

  const int tid  = threadIdx.x;
  const int w    = tid >> 5;
  const int lane = tid & 31;
  const int mn   = lane & 15;
  const int half = lane >> 4;
  const int kk   = half * 2;
  const int m0   = blockIdx.x * 16;
  const int n0   = w * 16;

  for (int idx = tid; idx < 16 * (F_INF / 4); idx += blockDim.x) {
    int row = idx >> 7;
    int c4  = idx & 127;
    float4 v = *(const float4*)&x[(size_t)(m0 + row) * F_INF + c4 * 4];
    float* d = &xs[row * XS_STRIDE + c4 * 4];
    d[0] = v.x; d[1] = v.y; d[2] = v.z; d[3] = v.w;
  }
  __syncthreads();

  v8f acc = {};
#pragma unroll 4
  for (int k0 = 0; k0 < F_INF; k0 += 4) {
    v2f av, bv;
    av.x = xs[mn * XS_STRIDE + k0 + kk];
    av.y = xs[mn * XS_STRIDE + k0 + kk + 1];
    bv.x = W1[(size_t)(k0 + kk) * HIDF + n0 + mn];
    bv.y = W1[(size_t)(k0 + kk + 1) * HIDF + n0 + mn];
    acc = wmma4(av, bv, acc);
  }

  float bias = b1[n0 + mn];
#pragma unroll
  for (int r = 0; r < 8; ++r) {
    int row = half ? r + 8 : r;
    float v = acc[r] + bias;
    hs[row * HS_STRIDE + n0 + mn] = v > 0.0f ? v : 0.0f;
  }
  __syncthreads();

  if (w == 0) {  // uniform branch -> EXEC all ones inside
    v8f zacc = {};
#pragma unroll
    for (int k0 = 0; k0 < HIDF; k0 += 4) {
      v2f av, bv;
      av.x = hs[mn * HS_STRIDE + k0 + kk];
      av.y = hs[mn * HS_STRIDE + k0 + kk + 1];
      bv.x = W2[(k0 + kk) * CCH + mn];
      bv.y = W2[(k0 + kk + 1) * CCH + mn];
      zacc = wmma4(av, bv, zacc);
    }
    float bb = b2[mn];
#pragma unroll
    for (int r = 0; r < 8; ++r) {
      int row = half ? r + 8 : r;
      zb[(size_t)(m0 + row) * CCH + mn] = zacc[r] + bb;
    }
  }
}

// ---------------------------------------------------------------------------
// Kernel 3: t += (z^T @ U) over one j-chunk, K-split 16-ways for concurrency
// (8192 waves in flight -> enough outstanding loads to saturate 23.3 TB/s).
// All 8 waves of a block share one j-chunk: z slice staged once in LDS.
// Partials accumulated with hardware fp32 atomics into tbuf.
// ---------------------------------------------------------------------------
__global__ void utz_kernel(const float* __restrict__ U,
                           const float* __restrict__ zb,
                           float* __restrict__ tbuf) {
  __shared__ float zs[JCHUNK * CCH];   // 32 KB

  const int tid  = threadIdx.x;
  const int lane = tid & 31;
  const int mn   = lane & 15;
  const int half = lane >> 4;
  const int kk   = half * 2;
  const int W    = blockIdx.x * 8 + (tid >> 5);
  const int i0   = (W & 511) * 16;        // i-tile
  const int jb   = (W >> 9) * JCHUNK;     // j-chunk base (same for whole block)

  // Stage z[jb:jb+512, :] into LDS (2048 float4, 8 per thread, coalesced).
  for (int idx = tid; idx < JCHUNK * CCH / 4; idx += blockDim.x) {
    float4 v = *(const float4*)&zb[(size_t)jb * CCH + idx * 4];
    float* d = &zs[idx * 4];
    d[0] = v.x; d[1] = v.y; d[2] = v.z; d[3] = v.w;
  }
  __syncthreads();

  v8f acc = {};
#pragma unroll 4
  for (int jl = 0; jl < JCHUNK; jl += 4) {
    v2f av, bv;
    av.x = zs[(jl + kk) * CCH + mn];            // A = z^T : A[c][j] = z[j][c]
    av.y = zs[(jl + kk + 1) * CCH + mn];
    bv.x = U[(size_t)(jb + jl + kk) * N_NODES + i0 + mn];   // B = U rows
    bv.y = U[(size_t)(jb + jl + kk + 1) * N_NODES + i0 + mn];
    acc = wmma4(av, bv, acc);
  }
#pragma unroll
  for (int r = 0; r < 8; ++r) {
    int c = half ? r + 8 : r;
    unsafeAtomicAdd(&tbuf[(size_t)(i0 + mn) * CCH + c], acc[r]);
  }
}

// ---------------------------------------------------------------------------
// Kernel 4: obuf += U[:, kc] @ (Hf ⊙ t)[kc]  (K-split 16-ways).
// Block = 256 threads (8 waves) covers 128 out rows; U staged 128x64/chunk,
// Hf⊙t fused into the 64x16 B-tile staging.
// ---------------------------------------------------------------------------
#define UT_STRIDE 68   // 64 + 4 pad

__global__ void outgemm_kernel(const float* __restrict__ U,
                               const float* __restrict__ tbuf,
                               const float* __restrict__ Hf,
                               float* __restrict__ obuf) {
  __shared__ float Ut[128 * UT_STRIDE];   // ~34.8 KB
  __shared__ float St[64 * CCH];          // 4 KB

  const int tid  = threadIdx.x;
  const int w    = tid >> 5;
  const int lane = tid & 31;
  const int mn   = lane & 15;
  const int half = lane >> 4;
  const int kk   = half * 2;
  const int r0   = (blockIdx.x & 63) * 128;        // out-row block
  const int kb   = (blockIdx.x >> 6) * JCHUNK;     // k-chunk base

  v8f acc = {};
  for (int i0 = kb; i0 < kb + JCHUNK; i0 += 64) {
    for (int idx = tid; idx < 128 * 16; idx += blockDim.x) {
      int row = idx >> 4;
      int c4  = idx & 15;
      float4 v = *(const float4*)&U[(size_t)(r0 + row) * N_NODES + i0 + c4 * 4];
      float* d = &Ut[row * UT_STRIDE + c4 * 4];
      d[0] = v.x; d[1] = v.y; d[2] = v.z; d[3] = v.w;
    }
    {  // B tile: s = Hf ⊙ t, fused at staging time (one float4 per thread)
      size_t base = (size_t)(i0 + (tid >> 2)) * CCH + (tid & 3) * 4;
      float4 tv = *(const float4*)&tbuf[base];
      float4 hv = *(const float4*)&Hf[base];
      float* d = &St[(tid >> 2) * CCH + (tid & 3) * 4];
      d[0] = tv.x * hv.x; d[1] = tv.y * hv.y; d[2] = tv.z * hv.z; d[3] = tv.w * hv.w;
    }
    __syncthreads();
#pragma unroll
    for (int k0 = 0; k0 < 64; k0 += 4) {
      v2f av, bv;
      av.x = Ut[(16 * w + mn) * UT_STRIDE + k0 + kk];
      av.y = Ut[(16 * w + mn) * UT_STRIDE + k0 + kk + 1];
      bv.x = St[(k0 + kk) * CCH + mn];
      bv.y = St[(k0 + kk + 1) * CCH + mn];
      acc = wmma4(av, bv, acc);
    }
    __syncthreads();
  }
#pragma unroll
  for (int r = 0; r < 8; ++r) {
    int row = half ? r + 8 : r;
    unsafeAtomicAdd(&obuf[(size_t)(r0 + 16 * w + row) * CCH + mn], acc[r]);
  }
}

// ---------------------------------------------------------------------------
// Kernel 5: row-wise log_softmax over obuf -> d_out.  One row per thread.
// ---------------------------------------------------------------------------
__global__ void lsm_kernel(const float* __restrict__ obuf,
                           float* __restrict__ out) {
  int n = blockIdx.x * blockDim.x + threadIdx.x;
  if (n >= N_NODES) return;
  float v[CCH];
#pragma unroll
  for (int q = 0; q < 4; ++q) {
    float4 t = *(const float4*)&obuf[(size_t)n * CCH + q * 4];
    v[q * 4 + 0] = t.x; v[q * 4 + 1] = t.y; v[q * 4 + 2] = t.z; v[q * 4 + 3] = t.w;
  }
  float mx = v[0];
#pragma unroll
  for (int i = 1; i < CCH; ++i) mx = fmaxf(mx, v[i]);
  float sm = 0.0f;
#pragma unroll
  for (int i = 0; i < CCH; ++i) sm += expf(v[i] - mx);
  float lse = mx + logf(sm);
#pragma unroll
  for (int q = 0; q < 4; ++q) {
    float4 t;
    t.x = v[q * 4 + 0] - lse; t.y = v[q * 4 + 1] - lse;
    t.z = v[q * 4 + 2] - lse; t.w = v[q * 4 + 3] - lse;
    *(float4*)&out[(size_t)n * CCH + q * 4] = t;
  }
}

// ---------------------------------------------------------------------------
extern "C" void kernel_launch(void* const* d_in, const int* in_sizes, int n_in,
                              void* d_out, int out_size, void* d_ws, size_t ws_size,
                              hipStream_t stream) {
  const float* origin_e = (const float*)d_in[0];
  const float* U        = (const float*)d_in[1];
  const float* x        = (const float*)d_in[2];
  const float* W1       = (const float*)d_in[3];
  const float* b1       = (const float*)d_in[4];
  const float* W2       = (const float*)d_in[5];
  const float* b2       = (const float*)d_in[6];
  const float* temp     = (const float*)d_in[7];
  float* out = (float*)d_out;

  float* ws   = (float*)d_ws;
  float* Hf   = ws;                             // [8192,16]
  float* zb   = Hf   + (size_t)N_NODES * CCH;   // [8192,16]
  float* tbuf = zb   + (size_t)N_NODES * CCH;   // [8192,16]  (accumulator)
  float* obuf = tbuf + (size_t)N_NODES * CCH;   // [8192,16]  (accumulator)

  // zero tbuf+obuf (contiguous): 2*131072 floats = 65536 float4
  zero_kernel<<<256, 256, 0, stream>>>((float4*)tbuf);
  jacobi_kernel<<<N_NODES / 256, 256, 0, stream>>>(origin_e, temp, Hf);
  mlp_kernel<<<N_NODES / 16, 128, 0, stream>>>(x, W1, b1, W2, b2, zb);
  utz_kernel<<<(N_NODES / 16) * NCHUNKS / 8, 256, 0, stream>>>(U, zb, tbuf);
  outgemm_kernel<<<64 * NCHUNKS, 256, 0, stream>>>(U, tbuf, Hf, obuf);
  lsm_kernel<<<N_NODES / 256, 256, 0, stream>>>(obuf, out);
}